// MultiboxLoss_8873402434107
// MI455X (gfx1250) — compile-verified
//
#include <hip/hip_runtime.h>
#include <stdint.h>

#define BB 32
#define GG 32
#define NBA 8732
#define CC 81
#define NEGPOS 3

typedef float v2f __attribute__((ext_vector_type(2)));
typedef float v8f __attribute__((ext_vector_type(8)));
typedef unsigned int v4u __attribute__((ext_vector_type(4)));
typedef int v4i __attribute__((ext_vector_type(4)));
typedef int v8i __attribute__((ext_vector_type(8)));

// ---- workspace layout (bytes) ----
#define OFF_ENC 0                       // B*NB*4 f32  = 4,470,784
#define OFF_LAB 4470784                 // B*NB  i32   = 1,117,696
#define OFF_CE  5588480                 // B*NB  f32
#define OFF_CLS 6706176                 // B*NB  f32
#define OFF_NP  7823872                 // B     i32
#define OFF_ACC 7824000                 // locSum f32, clsSum f32, nposTot i32, indCnt i32

// LDS byte offset of a __shared__ object: flat addr low 32 bits == LDS offset on gfx1250.
static __device__ __forceinline__ unsigned lds_off(const void* p) {
  return (unsigned)(uintptr_t)p;
}

// 1-D contiguous TDM load: nelem dwords from gaddr -> LDS at lds_byte_addr.
static __device__ __forceinline__ void tdm_load_1d(unsigned lds_byte_addr,
                                                   const void* gaddr,
                                                   unsigned nelem_dw) {
  unsigned long long ga = (unsigned long long)(uintptr_t)gaddr;
  v4u g0;
  g0.x = 1u;                                            // count=1 (valid), gather off
  g0.y = lds_byte_addr;                                 // lds_addr [63:32]
  g0.z = (unsigned)(ga & 0xFFFFFFFFu);                  // global_addr lo
  g0.w = (unsigned)((ga >> 32) & 0x01FFFFFFu) | (2u << 30); // addr hi | type=2
  v8i g1;
  g1[0] = (int)(2u << 16);                              // data_size = 4B
  g1[1] = (int)((nelem_dw & 0xFFFFu) << 16);            // tensor_dim0 (lo16)
  g1[2] = (int)(((nelem_dw >> 16) & 0xFFFFu) | (1u << 16)); // tensor_dim0 hi | tensor_dim1=1 lo
  g1[3] = (int)((nelem_dw & 0xFFFFu) << 16);            // tensor_dim1 hi=0 | tile_dim0 = nelem
  g1[4] = 0;                                            // tile_dim1=0 (unused), tile_dim2=0
  g1[5] = (int)nelem_dw;                                // tensor_dim0_stride (don't-care, 1D)
  g1[6] = 0;
  g1[7] = 0;
  v4i g2 = {0, 0, 0, 0};
  v4i g3 = {0, 0, 0, 0};
  v8i g4 = {0, 0, 0, 0, 0, 0, 0, 0};
  __builtin_amdgcn_tensor_load_to_lds(g0, g1, g2, g3, g4, 0);
}

// ---------------- K0: zero accumulators ----------------
__global__ void k0_init(float* locSum, float* clsSum, int* nposTot, int* indCnt) {
  if (threadIdx.x == 0) {
    *locSum = 0.f; *clsSum = 0.f; *nposTot = 0; *indCnt = 0;
  }
}

// ---------------- K1: per-image matching ----------------
__global__ __launch_bounds__(256) void k1_match(
    const float* __restrict__ anc, const float* __restrict__ gtb,
    const int* __restrict__ gtl, float* __restrict__ encG, int* __restrict__ labG,
    int* __restrict__ numPos, int* __restrict__ nposTot) {
  __shared__ float gx0[GG], gy0[GG], gx1[GG], gy1[GG], gA[GG];
  __shared__ float gcx[GG], gcy[GG], gw[GG], gh[GG];
  __shared__ int glab[GG];
  __shared__ unsigned long long bkey[GG];
  __shared__ float bestov[NBA];
  __shared__ unsigned char labL[NBA];
  __shared__ unsigned int cnt;

  const int b = blockIdx.x, t = threadIdx.x;
  if (t < GG) {
    const float* g = gtb + ((size_t)b * GG + t) * 4;
    float x = g[0], y = g[1], w = g[2], h = g[3];
    gcx[t] = x; gcy[t] = y; gw[t] = w; gh[t] = h;
    gx0[t] = x; gy0[t] = y; gx1[t] = x + w; gy1[t] = y + h;
    gA[t] = w * h;
    glab[t] = gtl[b * GG + t];
    bkey[t] = (unsigned long long)(NBA - 1);  // ov=0, idx=0 sentinel (matches argmax of all-zeros)
  }
  if (t == 0) cnt = 0u;
  __syncthreads();

  float bv[GG]; int bi[GG];
#pragma unroll
  for (int g = 0; g < GG; ++g) { bv[g] = -1.f; bi[g] = 0; }

  for (int i = t; i < NBA; i += 256) {
    float4 a = ((const float4*)anc)[i];
    float ax1 = a.x + a.z, ay1 = a.y + a.w;
    float areaB = a.z * a.w;
    float best = -1.f; int bg = 0;
#pragma unroll
    for (int g = 0; g < GG; ++g) {
      float lx = fmaxf(gx0[g], a.x), ly = fmaxf(gy0[g], a.y);
      float rx = fminf(gx1[g], ax1), ry = fminf(gy1[g], ay1);
      float iw = fmaxf(rx - lx, 0.f), ih = fmaxf(ry - ly, 0.f);
      float inter = iw * ih;
      float ov = inter / (gA[g] + areaB - inter);
      if (ov > best) { best = ov; bg = g; }      // argmax over gt (first wins)
      if (ov > bv[g]) { bv[g] = ov; bi[g] = i; } // argmax over anchors (first wins)
    }
    bestov[i] = best;
    int lab = 0;
    float4 e = make_float4(0.f, 0.f, 0.f, 0.f);
    if (best > 0.5f) {
      lab = glab[bg];
      e.x = (gcx[bg] - a.x) / a.z;
      e.y = (gcy[bg] - a.y) / a.w;
      e.z = __logf(gw[bg]) - __logf(a.z);
      e.w = __logf(gh[bg]) - __logf(a.w);
    }
    ((float4*)encG)[(size_t)b * NBA + i] = e;
    labL[i] = (unsigned char)lab;
  }
  // per-gt argmax reduce: pack (ov_bits, NBA-1-idx) so max == first-max semantics
#pragma unroll
  for (int g = 0; g < GG; ++g) {
    if (bv[g] >= 0.f) {
      unsigned long long key =
          ((unsigned long long)__float_as_uint(bv[g]) << 32) |
          (unsigned)(NBA - 1 - bi[g]);
      atomicMax(&bkey[g], key);
    }
  }
  __syncthreads();

  // scatter: best prior per gt (only where thr[j] is false — row2 wins otherwise)
  if (t < GG) {
    int g = t;
    int j = NBA - 1 - (int)(bkey[g] & 0xFFFFFFFFu);
    if (bestov[j] <= 0.5f) {
      float4 a = ((const float4*)anc)[j];
      float4 e;
      e.x = (gcx[g] - a.x) / a.z;
      e.y = (gcy[g] - a.y) / a.w;
      e.z = __logf(gw[g]) - __logf(a.z);
      e.w = __logf(gh[g]) - __logf(a.w);
      ((float4*)encG)[(size_t)b * NBA + j] = e;
      labL[j] = (unsigned char)glab[g];
    }
  }
  __syncthreads();

  unsigned int c = 0;
  for (int i = t; i < NBA; i += 256) {
    int l = (int)labL[i];
    labG[(size_t)b * NBA + i] = l;
    c += (l > 0) ? 1u : 0u;
  }
  atomicAdd(&cnt, c);
  __syncthreads();
  if (t == 0) { numPos[b] = (int)cnt; atomicAdd(nposTot, (int)cnt); }
}

// ---------------- K2: TDM-staged logsumexp (WMMA sum-of-exp) + smooth-L1 ----------------
__global__ __launch_bounds__(128) void k2_lse(
    const float* __restrict__ conf, const float* __restrict__ loc,
    const float* __restrict__ encG, const int* __restrict__ labG,
    float* __restrict__ ceG, float* __restrict__ clsG, float* __restrict__ locSum) {
  __shared__ float tile[128 * CC];
  __shared__ float maxv[128];
  __shared__ float sarr[128];

  const int b = blockIdx.y;
  const int r0 = blockIdx.x * 128;
  int n = NBA - r0; if (n > 128) n = 128;
  const int t = threadIdx.x;

  // Stage conf tile (n rows x 81 f32, contiguous) via Tensor Data Mover.
  const float* src = conf + ((size_t)b * NBA + r0) * CC;
  if (t < 32) {  // wave-uniform: wave 0 issues one TDM op
    tdm_load_1d(lds_off(tile), (const void*)src, (unsigned)(n * CC));
    __builtin_amdgcn_s_wait_tensorcnt(0);
  }
  __syncthreads();

  // Phase 1 (thread-per-anchor): row max, target logit, then exp(x - max) IN PLACE.
  // Tail rows (t >= n) are zero-filled so the WMMA phase needs no guards at all.
  int lab = 0; float xlab = 0.f;
  {
    float* row = &tile[t * CC];
    if (t < n) {
      float m = row[0];
      for (int c = 1; c < CC; ++c) m = fmaxf(m, row[c]);
      maxv[t] = m;
      lab = labG[(size_t)b * NBA + r0 + t];
      xlab = row[lab];
      for (int c = 0; c < CC; ++c) row[c] = __expf(row[c] - m);
    } else {
      for (int c = 0; c < CC; ++c) row[c] = 0.f;
    }
  }
  __syncthreads();

  // Phase 2: WMMA sum-of-exp. D = ones(16x4) x B(4x16), accumulated over k-chunks.
  // All LDS reads are unconditional and in-bounds -> no exec-mask branches.
  const int wave = t >> 5, lane = t & 31;
  const int koff = (lane >= 16) ? 2 : 0;   // hi half-wave feeds K=2,3 rows of B
  const bool lo = lane < 16;
  v2f A; A.x = 1.0f; A.y = 1.0f;
#pragma unroll
  for (int h = 0; h < 2; ++h) {
    const int arow = wave * 32 + h * 16 + (lane & 15);
    const float* row = &tile[arow * CC];
    v8f acc = {};
#pragma unroll
    for (int kb = 0; kb < 80; kb += 4) {   // 20 chunks, k <= 79: always in-bounds
      v2f Bv;
      Bv.x = row[kb + koff];
      Bv.y = row[kb + koff + 1];
      acc = __builtin_amdgcn_wmma_f32_16x16x4_f32(false, A, false, Bv,
                                                  (short)0, acc, false, false);
    }
    {  // final chunk: class 80 on lo lanes only; value-select, not a branch
      float v80 = row[80];
      v2f Bv;
      Bv.x = lo ? v80 : 0.f;
      Bv.y = 0.f;
      acc = __builtin_amdgcn_wmma_f32_16x16x4_f32(false, A, false, Bv,
                                                  (short)0, acc, false, false);
    }
    if (lo) sarr[arow] = acc[0];  // D[0, n] = sum_k B[k, n]
  }
  __syncthreads();

  if (t < n) {
    const size_t idx = (size_t)b * NBA + r0 + t;
    float lse = maxv[t] + __logf(sarr[t]);
    float cef = lse - xlab;                  // full CE for this row
    ceG[idx] = cef;
    clsG[idx] = (lab > 0) ? 0.f : cef;       // mining loss (zero on positives)
    if (lab > 0) {                           // smooth-L1 on positives
      const float* lp = loc + idx * 4;
      const float* ep = encG + idx * 4;
      float s = 0.f;
#pragma unroll
      for (int k = 0; k < 4; ++k) {
        float d = fabsf(lp[k] - ep[k]);
        s += (d < 1.f) ? 0.5f * d * d : d - 0.5f;
      }
      atomicAdd(locSum, s);
    }
  }
}

// ---------------- K3: per-image top-k selection (3-level radix histogram) ----------------
__global__ __launch_bounds__(256) void k3_select(
    const float* __restrict__ clsG, const float* __restrict__ ceG,
    const int* __restrict__ labG, const int* __restrict__ numPos,
    float* __restrict__ clsSum, int* __restrict__ indCnt) {
  __shared__ float vals[NBA];
  __shared__ unsigned int hist[2048];
  __shared__ unsigned char flg[NBA];
  __shared__ unsigned int sel1, sel2, above_s, tvbits, rrem;
  __shared__ float fsum;
  __shared__ unsigned int usum;

  const int b = blockIdx.x, t = threadIdx.x;
  for (int i = t; i < NBA; i += 256) vals[i] = clsG[(size_t)b * NBA + i];
  const int np = numPos[b];
  int nneg = NEGPOS * np; if (nneg > NBA - 1) nneg = NBA - 1;
  if (t == 0) { fsum = 0.f; usum = 0u; }
  __syncthreads();

  unsigned tv = 0xFFFFFFFFu;  // sentinel: nothing is > tv
  unsigned r3 = 0;
  if (nneg > 0) {
    // level 1: bits [31:21]
    for (int i = t; i < 2048; i += 256) hist[i] = 0u;
    __syncthreads();
    for (int i = t; i < NBA; i += 256)
      atomicAdd(&hist[__float_as_uint(vals[i]) >> 21], 1u);
    __syncthreads();
    if (t == 0) {
      unsigned cum = 0; int kp = 0;
      for (int k = 2047; k >= 0; --k) {
        if (cum + hist[k] >= (unsigned)nneg) { kp = k; break; }
        cum += hist[k];
      }
      sel1 = (unsigned)kp; above_s = cum;
    }
    __syncthreads();
    const unsigned s1 = sel1;
    // level 2: bits [20:10]
    for (int i = t; i < 2048; i += 256) hist[i] = 0u;
    __syncthreads();
    for (int i = t; i < NBA; i += 256) {
      unsigned bt = __float_as_uint(vals[i]);
      if ((bt >> 21) == s1) atomicAdd(&hist[(bt >> 10) & 0x7FFu], 1u);
    }
    __syncthreads();
    if (t == 0) {
      unsigned r = (unsigned)nneg - above_s, cum = 0; int kp = 0;
      for (int k = 2047; k >= 0; --k) {
        if (cum + hist[k] >= r) { kp = k; break; }
        cum += hist[k];
      }
      sel2 = (unsigned)kp; above_s += cum;
    }
    __syncthreads();
    const unsigned pref = (s1 << 11) | sel2;
    // level 3: bits [9:0]
    for (int i = t; i < 2048; i += 256) hist[i] = 0u;
    __syncthreads();
    for (int i = t; i < NBA; i += 256) {
      unsigned bt = __float_as_uint(vals[i]);
      if ((bt >> 10) == pref) atomicAdd(&hist[bt & 0x3FFu], 1u);
    }
    __syncthreads();
    if (t == 0) {
      unsigned r = (unsigned)nneg - above_s, cum = 0; int kp = 0;
      for (int k = 1023; k >= 0; --k) {
        if (cum + hist[k] >= r) { kp = k; break; }
        cum += hist[k];
      }
      tvbits = (pref << 10) | (unsigned)kp;
      rrem = r - cum;  // how many ==threshold to take (lowest index first)
    }
    __syncthreads();
    tv = tvbits; r3 = rrem;
  }

  for (int i = t; i < NBA; i += 256)
    flg[i] = (nneg > 0 && __float_as_uint(vals[i]) > tv) ? 1 : 0;
  __syncthreads();
  if (t == 0 && r3 > 0) {  // stable tie-break: smallest indices among equals
    unsigned taken = 0;
    for (int i = 0; i < NBA; ++i) {
      if (__float_as_uint(vals[i]) == tv) {
        flg[i] = 1;
        if (++taken == r3) break;
      }
    }
  }
  __syncthreads();

  float sc = 0.f; unsigned cc = 0;
  for (int i = t; i < NBA; i += 256) {
    int lab = labG[(size_t)b * NBA + i];
    bool ind = (lab > 0) || flg[i];
    if (ind) { sc += ceG[(size_t)b * NBA + i]; cc++; }
  }
  atomicAdd(&fsum, sc);
  atomicAdd(&usum, cc);
  __syncthreads();
  if (t == 0) { atomicAdd(clsSum, fsum); atomicAdd(indCnt, (int)usum); }
}

// ---------------- K4: finalize scalar ----------------
__global__ void k4_final(const float* locSum, const float* clsSum,
                         const int* nposTot, const int* indCnt, float* out) {
  if (threadIdx.x == 0) {
    float Nf = (float)(*nposTot);
    // non-ind rows: logits all-zero -> CE = log(81) each
    float clsT = *clsSum + (float)(BB * NBA - *indCnt) * __logf(81.0f);
    out[0] = (*locSum) / Nf + clsT / Nf;
  }
}

extern "C" void kernel_launch(void* const* d_in, const int* in_sizes, int n_in,
                              void* d_out, int out_size, void* d_ws, size_t ws_size,
                              hipStream_t stream) {
  (void)in_sizes; (void)n_in; (void)out_size; (void)ws_size;
  const float* loc  = (const float*)d_in[0];
  const float* conf = (const float*)d_in[1];
  const float* anc  = (const float*)d_in[2];
  const float* gtb  = (const float*)d_in[3];
  const int*   gtl  = (const int*)d_in[4];

  char* ws = (char*)d_ws;
  float* encG   = (float*)(ws + OFF_ENC);
  int*   labG   = (int*)  (ws + OFF_LAB);
  float* ceG    = (float*)(ws + OFF_CE);
  float* clsG   = (float*)(ws + OFF_CLS);
  int*   numPos = (int*)  (ws + OFF_NP);
  float* locSum = (float*)(ws + OFF_ACC + 0);
  float* clsSum = (float*)(ws + OFF_ACC + 4);
  int*   nposT  = (int*)  (ws + OFF_ACC + 8);
  int*   indCnt = (int*)  (ws + OFF_ACC + 12);

  k0_init<<<1, 64, 0, stream>>>(locSum, clsSum, nposT, indCnt);
  k1_match<<<BB, 256, 0, stream>>>(anc, gtb, gtl, encG, labG, numPos, nposT);
  dim3 g2((NBA + 127) / 128, BB);
  k2_lse<<<g2, 128, 0, stream>>>(conf, loc, encG, labG, ceG, clsG, locSum);
  k3_select<<<BB, 256, 0, stream>>>(clsG, ceG, labG, numPos, clsSum, indCnt);
  k4_final<<<1, 32, 0, stream>>>(locSum, clsSum, nposT, indCnt, (float*)d_out);
}